// SO3net_13597866459346
// MI455X (gfx1250) — compile-verified
//
#include <hip/hip_runtime.h>
#include <math.h>

// SO3net forward for MI455X (gfx1250, wave32).
// Per-atom 32x32 matmuls on the f32 WMMA pipe (v_wmma_f32_16x16x4_f32).
// Edge aggregation: CSR rowptr from sorted idx_i -> one wave per atom,
// lane = feature channel, no atomics (deterministic).
// CG contraction uses compile-time structural sparsity (selection rules)
// with device-computed values; LDS tables padded to odd strides to kill
// bank conflicts on the 64-bank LDS.

#define NB    32
#define MSH   9
#define NL    3
#define NRBF  20
#define TSTEPS 2
#define FCUT  5.0f

#define WFP   (NRBF + 1)   // padded row stride for Wf in LDS (21, odd)
#define WGP_  (NB + 1)     // padded row stride for Wg in LDS (33, odd)

typedef float v2f __attribute__((ext_vector_type(2)));
typedef float v8f __attribute__((ext_vector_type(8)));

__device__ __forceinline__ int lidx_of(int c) { return (c >= 4) ? 2 : ((c >= 1) ? 1 : 0); }

// Structural-nonzero predicate for the real-SH CG tensor (lmax=2).
// Selection rules: parity, triangle, even # of sine-type (m<0) members,
// |m3| in {|m1|+|m2|, ||m1|-|m2||}.  Constant-folds inside unrolled loops.
__device__ static constexpr bool cg_nonzero(int a, int b, int c) {
    int l1 = (a >= 4) ? 2 : (a >= 1 ? 1 : 0); int m1 = a - l1 * l1 - l1;
    int l2 = (b >= 4) ? 2 : (b >= 1 ? 1 : 0); int m2 = b - l2 * l2 - l2;
    int l3 = (c >= 4) ? 2 : (c >= 1 ? 1 : 0); int m3 = c - l3 * l3 - l3;
    int ld = (l1 > l2) ? (l1 - l2) : (l2 - l1);
    if (((l1 + l2 + l3) & 1) || l3 < ld || l3 > l1 + l2) return false;
    int neg = (m1 < 0 ? 1 : 0) + (m2 < 0 ? 1 : 0) + (m3 < 0 ? 1 : 0);
    if (neg & 1) return false;
    int am1 = m1 < 0 ? -m1 : m1, am2 = m2 < 0 ? -m2 : m2, am3 = m3 < 0 ? -m3 : m3;
    int sum = am1 + am2, dif = (am1 > am2) ? (am1 - am2) : (am2 - am1);
    return (am3 == sum) || (am3 == dif);
}

// ---------------------------------------------------------------------------
// CG table construction (device, fp64) — real-SH Clebsch-Gordan for lmax=2.
// ---------------------------------------------------------------------------
struct C2 { double re, im; };

__device__ double dfact(int k) { double f = 1.0; for (int i = 2; i <= k; ++i) f *= (double)i; return f; }

__device__ double cg_complex(int j1,int m1,int j2,int m2,int j3,int m3) {
    if (m3 != m1 + m2 || j3 < abs(j1 - j2) || j3 > j1 + j2) return 0.0;
    double pref = sqrt((2.0*j3+1.0) * dfact(j1+j2-j3) * dfact(j1-j2+j3) * dfact(-j1+j2+j3) / dfact(j1+j2+j3+1));
    pref *= sqrt(dfact(j1+m1)*dfact(j1-m1)*dfact(j2+m2)*dfact(j2-m2)*dfact(j3+m3)*dfact(j3-m3));
    int kmin = 0;
    if (j2 - j3 - m1 > kmin) kmin = j2 - j3 - m1;
    if (j1 + m2 - j3 > kmin) kmin = j1 + m2 - j3;
    int kmax = j1 + j2 - j3;
    if (j1 - m1 < kmax) kmax = j1 - m1;
    if (j2 + m2 < kmax) kmax = j2 + m2;
    double s = 0.0;
    for (int k = kmin; k <= kmax; ++k) {
        double den = dfact(k)*dfact(j1+j2-j3-k)*dfact(j1-m1-k)*dfact(j2+m2-k)*dfact(j3-j2+m1+k)*dfact(j3-j1-m2+k);
        s += ((k & 1) ? -1.0 : 1.0) / den;
    }
    return pref * s;
}

__device__ int urow(int /*l*/, int m, int* mcol, C2* val) {
    const double s = 0.70710678118654752440;
    if (m == 0) { mcol[0] = 0; val[0] = {1.0, 0.0}; return 1; }
    if (m > 0) {
        mcol[0] =  m; val[0] = {((m & 1) ? -s : s), 0.0};
        mcol[1] = -m; val[1] = {s, 0.0};
        return 2;
    }
    int mu = -m;
    mcol[0] =  m; val[0] = {0.0, s};                    // i/sqrt(2)
    mcol[1] = -m; val[1] = {0.0, ((mu & 1) ? s : -s)};  // -i*(-1)^mu/sqrt(2)
    return 2;
}

__global__ void build_cg(float* __restrict__ CGout) {
    int t = blockIdx.x * blockDim.x + threadIdx.x;
    if (t >= 729) return;
    int a = t / 81, b = (t / 9) % 9, c = t % 9;
    int l1 = (a >= 4) ? 2 : (a >= 1 ? 1 : 0); int m1 = a - l1*l1 - l1;
    int l2 = (b >= 4) ? 2 : (b >= 1 ? 1 : 0); int m2 = b - l2*l2 - l2;
    int l3 = (c >= 4) ? 2 : (c >= 1 ? 1 : 0); int m3 = c - l3*l3 - l3;
    float out = 0.0f;
    if ((((l1 + l2 + l3) & 1) == 0) && (l3 >= abs(l1 - l2)) && (l3 <= l1 + l2)) {
        int ma[2], mb[2], mc[2]; C2 va[2], vb[2], vc[2];
        int na = urow(l1, m1, ma, va);
        int nb = urow(l2, m2, mb, vb);
        int nc = urow(l3, m3, mc, vc);
        double re = 0.0;
        for (int i = 0; i < na; ++i)
            for (int j = 0; j < nb; ++j)
                for (int k = 0; k < nc; ++k) {
                    double cg = cg_complex(l1, ma[i], l2, mb[j], l3, mc[k]);
                    if (cg == 0.0) continue;
                    double pr = va[i].re*vb[j].re - va[i].im*vb[j].im;
                    double pi = va[i].re*vb[j].im + va[i].im*vb[j].re;
                    re += (pr*vc[k].re + pi*vc[k].im) * cg;   // Re(va*vb*conj(vc))*cg
                }
        out = (float)re;
    }
    CGout[t] = out;
}

// ---------------------------------------------------------------------------
// CSR rowptr from sorted idx_i.
// ---------------------------------------------------------------------------
__global__ void build_rowptr(const int* __restrict__ idx_i, int* __restrict__ rowptr,
                             int nedges, int natoms) {
    int e = blockIdx.x * blockDim.x + threadIdx.x;
    if (e >= nedges) return;
    int i = idx_i[e];
    int prev = (e == 0) ? -1 : idx_i[e - 1];
    for (int a = prev + 1; a <= i; ++a) rowptr[a] = e;
    if (e == nedges - 1)
        for (int a = i + 1; a <= natoms; ++a) rowptr[a] = nedges;
}

// ---------------------------------------------------------------------------
// x init: l=0 row from embedding, rest zero.
// ---------------------------------------------------------------------------
__global__ void init_x(const int* __restrict__ zn, const float* __restrict__ emb,
                       float* __restrict__ x, int natoms) {
    int t = blockIdx.x * blockDim.x + threadIdx.x;
    if (t >= natoms * NB) return;
    int i = t / NB, n = t % NB;
    float* xp = x + (size_t)i * (MSH * NB);
    xp[n] = emb[zn[i] * NB + n];
#pragma unroll
    for (int c = 1; c < MSH; ++c) xp[c * NB + n] = 0.0f;
}

// ---------------------------------------------------------------------------
// Edge messages + segment sum.  One wave per atom, lane n = feature channel.
// ---------------------------------------------------------------------------
__global__ __launch_bounds__(256) void edge_msg(
    const float* __restrict__ r_ij, const int* __restrict__ idx_j,
    const int* __restrict__ rowptr, const float* __restrict__ x,
    const float* __restrict__ Wf_t, const float* __restrict__ bf_t,
    const float* __restrict__ CG, float* __restrict__ dxout, int natoms) {
    __shared__ float cg_l[729];
    __shared__ float wf_l[NL * NB * WFP];   // padded stride 21: conflict-free
    __shared__ float bf_l[NL * NB];
    for (int s = threadIdx.x; s < 729; s += 256) cg_l[s] = CG[s];
    for (int s = threadIdx.x; s < NL * NB * NRBF; s += 256) {
        int row = s / NRBF, k = s % NRBF;
        wf_l[row * WFP + k] = Wf_t[s];
    }
    for (int s = threadIdx.x; s < NL * NB; s += 256) bf_l[s] = bf_t[s];
    __syncthreads();

    const int lane = threadIdx.x & 31;
    const int wid  = threadIdx.x >> 5;
    const int i = blockIdx.x * 8 + wid;
    if (i >= natoms) return;

    float acc[MSH];
#pragma unroll
    for (int c = 0; c < MSH; ++c) acc[c] = 0.0f;

    const int e0 = rowptr[i], e1 = rowptr[i + 1];
    const float width = FCUT / (float)(NRBF - 1);
    const float gcoef = -0.5f / (width * width);

    for (int e = e0; e < e1; ++e) {
        int j = idx_j[e];
        if (e + 1 < e1) {  // prefetch next neighbor's feature row
            int jn = idx_j[e + 1];
            __builtin_prefetch(x + (size_t)jn * (MSH * NB) + lane, 0, 0);
        }
        float rx = r_ij[3 * e], ry = r_ij[3 * e + 1], rz = r_ij[3 * e + 2];
        float d2 = rx * rx + ry * ry + rz * rz;
        float rd = rsqrtf(d2);
        float d  = d2 * rd;
        float ux = rx * rd, uy = ry * rd, uz = rz * rd;

        float Y[MSH];
        const float c1 = 0.4886025119029199f, c2 = 1.0925484305920792f;
        Y[0] = 0.28209479177387814f;
        Y[1] = c1 * uy; Y[2] = c1 * uz; Y[3] = c1 * ux;
        Y[4] = c2 * ux * uy; Y[5] = c2 * uy * uz;
        Y[6] = 0.31539156525252005f * (3.0f * uz * uz - 1.0f);
        Y[7] = c2 * ux * uz;
        Y[8] = 0.5462742152960396f * (ux * ux - uy * uy);

        float fcut = (d < FCUT) ? 0.5f * (__cosf(d * 3.14159265358979323846f / FCUT) + 1.0f) : 0.0f;

        float radial[NRBF];
#pragma unroll
        for (int k = 0; k < NRBF; ++k) { float t = d - (float)k * width; radial[k] = __expf(gcoef * t * t); }

        float wl[NL];
#pragma unroll
        for (int l = 0; l < NL; ++l) {
            int row = l * NB + lane;
            float s = bf_l[row];
            const float* wr = &wf_l[row * WFP];
#pragma unroll
            for (int k = 0; k < NRBF; ++k) s += radial[k] * wr[k];
            wl[l] = fcut * s;
        }

        float xj[MSH];
        const float* xp = x + (size_t)j * (MSH * NB) + lane;
#pragma unroll
        for (int b = 0; b < MSH; ++b) xj[b] = xp[b * NB];

        float sc[MSH];
#pragma unroll
        for (int c = 0; c < MSH; ++c) sc[c] = 0.0f;
#pragma unroll
        for (int a = 0; a < MSH; ++a) {
            float ya = Y[a];
#pragma unroll
            for (int b = 0; b < MSH; ++b) {
                float p = ya * xj[b];   // dead-code-eliminated when no c survives
#pragma unroll
                for (int c = 0; c < MSH; ++c)
                    if (cg_nonzero(a, b, c)) sc[c] += cg_l[(a * 9 + b) * 9 + c] * p;
            }
        }
#pragma unroll
        for (int c = 0; c < MSH; ++c) acc[c] += wl[lidx_of(c)] * sc[c];
    }

    float* op = dxout + (size_t)i * (MSH * NB) + lane;
#pragma unroll
    for (int c = 0; c < MSH; ++c) op[c * NB] = acc[c];
}

// ---------------------------------------------------------------------------
// Y(Mtot,32) = X(Mtot,32) @ W(32,32)^T via v_wmma_f32_16x16x4_f32.
// A: lane L<16 -> (M=L, K=0/1) in v0/v1; lane L+16 -> K=2/3.
// B: lane L<16 -> (K=0/1, N=L);      lane L+16 -> (K=2/3, N=L).
// D: vgpr r, lanes 0-15 -> (M=r, N=lane); lanes 16-31 -> (M=8+r, N=lane-16).
// ---------------------------------------------------------------------------
__global__ __launch_bounds__(256) void gemm_nt_wmma(
    const float* __restrict__ X, const float* __restrict__ W,
    float* __restrict__ Y, int Mtot, int accumulate) {
    const int lane = threadIdx.x & 31;
    const int w = (blockIdx.x * 256 + threadIdx.x) >> 5;
    const int row0 = w * 16;
    if (row0 >= Mtot) return;
    const int l16 = lane & 15, lh = lane >> 4;

    int rload = row0 + l16;
    if (rload > Mtot - 1) rload = Mtot - 1;

    v8f acc0 = {}; v8f acc1 = {};
    const float* xr = X + (size_t)rload * NB + 2 * lh;
    const float* w0 = W + (size_t)l16 * NB + 2 * lh;
    const float* w1 = W + (size_t)(16 + l16) * NB + 2 * lh;
#pragma unroll
    for (int s = 0; s < 8; ++s) {
        v2f a, b0, b1;
        a.x  = xr[s * 4];  a.y  = xr[s * 4 + 1];
        b0.x = w0[s * 4];  b0.y = w0[s * 4 + 1];
        b1.x = w1[s * 4];  b1.y = w1[s * 4 + 1];
        acc0 = __builtin_amdgcn_wmma_f32_16x16x4_f32(false, a, false, b0, (short)0, acc0, false, false);
        acc1 = __builtin_amdgcn_wmma_f32_16x16x4_f32(false, a, false, b1, (short)0, acc1, false, false);
    }
#pragma unroll
    for (int r = 0; r < 8; ++r) {
        int m = row0 + r + 8 * lh;
        if (m < Mtot) {
            float* y0 = Y + (size_t)m * NB + l16;
            if (accumulate) { y0[0] += acc0[r]; y0[16] += acc1[r]; }
            else            { y0[0]  = acc0[r]; y0[16]  = acc1[r]; }
        }
    }
}

// ---------------------------------------------------------------------------
// dx[c,n] += sum_ab CG[a,b,c] * dx[a,n] * ddx[b,n]   (one wave per atom)
// ---------------------------------------------------------------------------
__global__ __launch_bounds__(256) void cgmix(
    float* __restrict__ dx, const float* __restrict__ ddx,
    const float* __restrict__ CG, int natoms) {
    __shared__ float cg_l[729];
    for (int s = threadIdx.x; s < 729; s += 256) cg_l[s] = CG[s];
    __syncthreads();
    int lane = threadIdx.x & 31, wid = threadIdx.x >> 5;
    int i = blockIdx.x * 8 + wid;
    if (i >= natoms) return;
    float da[MSH], db[MSH], out[MSH];
    float* dp = dx + (size_t)i * (MSH * NB) + lane;
    const float* bp = ddx + (size_t)i * (MSH * NB) + lane;
#pragma unroll
    for (int c = 0; c < MSH; ++c) { da[c] = dp[c * NB]; db[c] = bp[c * NB]; out[c] = da[c]; }
#pragma unroll
    for (int a = 0; a < MSH; ++a)
#pragma unroll
        for (int b = 0; b < MSH; ++b) {
            float p = da[a] * db[b];
#pragma unroll
            for (int c = 0; c < MSH; ++c)
                if (cg_nonzero(a, b, c)) out[c] += cg_l[(a * 9 + b) * 9 + c] * p;
        }
#pragma unroll
    for (int c = 0; c < MSH; ++c) dp[c * NB] = out[c];
}

// ---------------------------------------------------------------------------
// h = sigmoid(dx[:,0,:] @ Wg^T + bg); dx[c] *= h[lidx(c)]
// ---------------------------------------------------------------------------
__global__ __launch_bounds__(256) void gate(
    float* __restrict__ dx, const float* __restrict__ Wg_t,
    const float* __restrict__ bg_t, int natoms) {
    __shared__ float wg_l[NL * NB * WGP_];   // padded stride 33: conflict-free
    __shared__ float bg_l[NL * NB];
    __shared__ float d0[8 * NB];
    for (int s = threadIdx.x; s < NL * NB * NB; s += 256) {
        int row = s / NB, k = s % NB;
        wg_l[row * WGP_ + k] = Wg_t[s];
    }
    for (int s = threadIdx.x; s < NL * NB; s += 256) bg_l[s] = bg_t[s];
    __syncthreads();
    int lane = threadIdx.x & 31, wid = threadIdx.x >> 5;
    int i = blockIdx.x * 8 + wid;
    float* dp = dx + (size_t)i * (MSH * NB) + lane;
    if (i < natoms) d0[wid * NB + lane] = dp[0];
    __syncthreads();
    if (i >= natoms) return;
    float h[NL];
#pragma unroll
    for (int l = 0; l < NL; ++l) {
        int j = l * NB + lane;
        float s = bg_l[j];
        const float* wr = &wg_l[j * WGP_];
        const float* xv = &d0[wid * NB];
#pragma unroll
        for (int k = 0; k < NB; ++k) s += xv[k] * wr[k];
        h[l] = 1.0f / (1.0f + __expf(-s));
    }
#pragma unroll
    for (int c = 0; c < MSH; ++c) dp[c * NB] *= h[lidx_of(c)];
}

// ---------------------------------------------------------------------------
extern "C" void kernel_launch(void* const* d_in, const int* in_sizes, int n_in,
                              void* d_out, int out_size, void* d_ws, size_t ws_size,
                              hipStream_t stream) {
    const int*   zn    = (const int*)d_in[0];
    const float* rij   = (const float*)d_in[1];
    const int*   idx_i = (const int*)d_in[2];
    const int*   idx_j = (const int*)d_in[3];
    const float* emb   = (const float*)d_in[4];
    const float* Wf    = (const float*)d_in[5];
    const float* bf    = (const float*)d_in[6];
    const float* Wm1   = (const float*)d_in[7];
    const float* Wm2   = (const float*)d_in[8];
    const float* Wm3   = (const float*)d_in[9];
    const float* Wg    = (const float*)d_in[10];
    const float* bg    = (const float*)d_in[11];
    float* x = (float*)d_out;

    const int natoms = in_sizes[0];
    const int nedges = in_sizes[1] / 3;
    const int Mtot   = natoms * MSH;

    // Workspace layout (~46 MB for N=20000): CG | rowptr | dx | ddx
    char* ws = (char*)d_ws;
    size_t off = 0;
    float* CG = (float*)(ws + off); off += 1024 * sizeof(float);
    int* rowptr = (int*)(ws + off); off += ((size_t)natoms + 2) * sizeof(int);
    off = (off + 255) & ~(size_t)255;
    float* dxb = (float*)(ws + off); off += (size_t)Mtot * NB * sizeof(float);
    float* ddx = (float*)(ws + off);

    build_cg<<<3, 256, 0, stream>>>(CG);
    build_rowptr<<<(nedges + 255) / 256, 256, 0, stream>>>(idx_i, rowptr, nedges, natoms);
    init_x<<<(natoms * NB + 255) / 256, 256, 0, stream>>>(zn, emb, x, natoms);

    const int ablocks = (natoms + 7) / 8;
    const int gwaves  = (Mtot + 15) / 16;
    const int gblocks = (gwaves * 32 + 255) / 256;

    for (int t = 0; t < TSTEPS; ++t) {
        edge_msg<<<ablocks, 256, 0, stream>>>(rij, idx_j, rowptr, x,
                                              Wf + (size_t)t * NL * NB * NRBF,
                                              bf + (size_t)t * NL * NB, CG, dxb, natoms);
        gemm_nt_wmma<<<gblocks, 256, 0, stream>>>(dxb, Wm1 + (size_t)t * NB * NB, ddx, Mtot, 0);
        cgmix<<<ablocks, 256, 0, stream>>>(dxb, ddx, CG, natoms);
        gemm_nt_wmma<<<gblocks, 256, 0, stream>>>(dxb, Wm2 + (size_t)t * NB * NB, dxb, Mtot, 0);
        gate<<<ablocks, 256, 0, stream>>>(dxb, Wg + (size_t)t * NL * NB * NB,
                                          bg + (size_t)t * NL * NB, natoms);
        gemm_nt_wmma<<<gblocks, 256, 0, stream>>>(dxb, Wm3 + (size_t)t * NB * NB, x, Mtot, 1);
    }
}